// KMeansSegmentation_30459908063373
// MI455X (gfx1250) — compile-verified
//
#include <hip/hip_runtime.h>

#define HW        262144      // 512*512
#define NBATCH    32
#define NCLUST    5
#define FIXSCALE  1048576.0f  // 2^20 fixed-point scale: deterministic integer reduction

typedef __attribute__((ext_vector_type(2))) float v2f;
typedef __attribute__((ext_vector_type(4))) float v4f;
typedef __attribute__((ext_vector_type(8))) float v8f;

// --- Init: deterministic centroids (linspace indices) + zero accumulators ---
__global__ void kmeans_init(const float* __restrict__ x, float* __restrict__ cent,
                            unsigned long long* __restrict__ accum) {
    const int t = threadIdx.x;
    if (t < NBATCH * NCLUST) {
        const int idx5[NCLUST] = {0, 65535, 131071, 196607, 262143};
        const int b = t / NCLUST, k = t % NCLUST;
        const int p = idx5[k];
        const float* xb = x + (size_t)b * 3 * HW;
        float* c = cent + (size_t)t * 3;
        c[0] = xb[p];
        c[1] = xb[p + HW];
        c[2] = xb[p + 2 * HW];
    }
    for (int i = t; i < NBATCH * NCLUST * 4; i += blockDim.x) accum[i] = 0ULL;
}

// --- Assign pass ---
// A (16x4) = centroids (M=cluster, K=channel); B (4x16) = pixels^T (N=pixel).
// D layout: lane<16 -> d[v] = dot(cent_v, pixel_lane%16) => lane-local argmin over v=0..4.
// 64-pixel tiles: lane n holds pixels base+4n..4n+3 via B128 loads; WMMA j consumes
// element j of each lane's float4, so pixel base+4n+j's result lands on lane n.
template <bool WRITE_LABELS>
__global__ void kmeans_assign(const float* __restrict__ x, const float* __restrict__ cent,
                              unsigned long long* __restrict__ accum, float* __restrict__ out) {
    __shared__ unsigned long long sacc[NCLUST * 4];
    const int  b    = blockIdx.y;
    const int  lane = threadIdx.x & 31;
    const int  wav  = threadIdx.x >> 5;
    const int  m    = lane & 15;
    const bool lo   = (lane < 16);

    if (!WRITE_LABELS) {
        if (threadIdx.x < NCLUST * 4) sacc[threadIdx.x] = 0ULL;
        __syncthreads();
    }

    // Broadcast-load this batch's 5 centroids, build |c|^2 and the A fragment.
    float cr[NCLUST], cg[NCLUST], cbv[NCLUST], csq[NCLUST];
    const float* cp = cent + (size_t)b * NCLUST * 3;
    #pragma unroll
    for (int k = 0; k < NCLUST; ++k) {
        cr[k]  = cp[k * 3 + 0];
        cg[k]  = cp[k * 3 + 1];
        cbv[k] = cp[k * 3 + 2];
        csq[k] = cr[k] * cr[k] + cg[k] * cg[k] + cbv[k] * cbv[k];
    }
    // A 16x4: lanes 0-15 hold K=0,1 (r,g); lanes 16-31 hold K=2,3 (b,pad0). Rows >=5 are 0.
    float a0 = 0.0f, a1 = 0.0f, a2 = 0.0f;
    #pragma unroll
    for (int k = 0; k < NCLUST; ++k) {
        if (m == k) { a0 = cr[k]; a1 = cg[k]; a2 = cbv[k]; }
    }
    v2f afrag;
    afrag.x = lo ? a0 : a2;
    afrag.y = lo ? a1 : 0.0f;

    // Per-lane local accumulators (fixed order => deterministic).
    float lsum[NCLUST][3];
    float lcnt[NCLUST];
    #pragma unroll
    for (int k = 0; k < NCLUST; ++k) {
        lsum[k][0] = lsum[k][1] = lsum[k][2] = 0.0f;
        lcnt[k] = 0.0f;
    }

    const float* xb   = x + (size_t)b * 3 * HW;
    const int nTiles  = HW / 64;                         // 4096 tiles of 64 pixels
    const int wavesPB = blockDim.x >> 5;
    const int tstride = gridDim.x * wavesPB;
    for (int tile = blockIdx.x * wavesPB + wav; tile < nTiles; tile += tstride) {
        const int pbase = tile << 6;
        const int poff  = pbase + (m << 2);
        // One B128 load per channel plane: lane n covers pixels pbase+4n..4n+3.
        const v4f r4 = *(const v4f*)(xb + poff);
        const v4f g4 = *(const v4f*)(xb + poff + HW);
        const v4f b4 = *(const v4f*)(xb + poff + 2 * HW);

        // Four independent WMMAs (full EXEC), one per float4 element.
        v8f d[4];
        #pragma unroll
        for (int j = 0; j < 4; ++j) {
            v2f bfrag;
            bfrag.x = lo ? r4[j] : b4[j];
            bfrag.y = lo ? g4[j] : 0.0f;
            v8f acc = {};
            d[j] = __builtin_amdgcn_wmma_f32_16x16x4_f32(false, afrag, false, bfrag,
                                                         (short)0, acc, false, false);
        }

        if (lo) {
            v4f lab;
            #pragma unroll
            for (int j = 0; j < 4; ++j) {
                // argmin_k (|c_k|^2 - 2 p.c_k) == argmin_k ||p - c_k||^2 (first-min ties)
                float best = csq[0] - 2.0f * d[j][0];
                int   bk   = 0;
                #pragma unroll
                for (int k = 1; k < NCLUST; ++k) {
                    const float s = csq[k] - 2.0f * d[j][k];
                    if (s < best) { best = s; bk = k; }
                }
                if (WRITE_LABELS) {
                    lab[j] = (float)bk;
                } else {
                    #pragma unroll
                    for (int k = 0; k < NCLUST; ++k) {
                        const float sel = (k == bk) ? 1.0f : 0.0f;
                        lsum[k][0] += sel * r4[j];
                        lsum[k][1] += sel * g4[j];
                        lsum[k][2] += sel * b4[j];
                        lcnt[k]    += sel;
                    }
                }
            }
            if (WRITE_LABELS) {
                // One contiguous B128 store of 4 labels per lane.
                *(v4f*)(out + (size_t)b * HW + poff) = lab;
            }
        }
    }

    if (!WRITE_LABELS) {
        if (lo) {
            #pragma unroll
            for (int k = 0; k < NCLUST; ++k) {
                atomicAdd(&sacc[k * 4 + 0], (unsigned long long)(lsum[k][0] * FIXSCALE + 0.5f));
                atomicAdd(&sacc[k * 4 + 1], (unsigned long long)(lsum[k][1] * FIXSCALE + 0.5f));
                atomicAdd(&sacc[k * 4 + 2], (unsigned long long)(lsum[k][2] * FIXSCALE + 0.5f));
                atomicAdd(&sacc[k * 4 + 3], (unsigned long long)(lcnt[k] + 0.5f));
            }
        }
        __syncthreads();
        if (threadIdx.x < NCLUST * 4) {
            atomicAdd(&accum[(size_t)b * NCLUST * 4 + threadIdx.x], sacc[threadIdx.x]);
        }
    }
}

// --- Centroid update: cent = count>0 ? sum/count : cent ; re-zero accumulators ---
__global__ void kmeans_update(float* __restrict__ cent, unsigned long long* __restrict__ accum) {
    const int t = threadIdx.x;
    if (t >= NBATCH * NCLUST) return;
    unsigned long long* a = accum + (size_t)t * 4;
    const unsigned long long sr = a[0], sg = a[1], sb = a[2], cn = a[3];
    a[0] = a[1] = a[2] = a[3] = 0ULL;   // ready for next assign pass
    if (cn > 0ULL) {
        const double inv = 1.0 / ((double)cn * (double)FIXSCALE);
        float* c = cent + (size_t)t * 3;
        c[0] = (float)((double)sr * inv);
        c[1] = (float)((double)sg * inv);
        c[2] = (float)((double)sb * inv);
    }
}

extern "C" void kernel_launch(void* const* d_in, const int* in_sizes, int n_in,
                              void* d_out, int out_size, void* d_ws, size_t ws_size,
                              hipStream_t stream) {
    (void)in_sizes; (void)n_in; (void)out_size; (void)ws_size;
    const float* x = (const float*)d_in[0];
    float*       out = (float*)d_out;

    // ws layout: [0, 1920)    float cent[32][5][3]
    //            [2048, 7168) u64 accum[32][5][4]  (sumR,sumG,sumB,count) fixed-point
    float*              cent  = (float*)d_ws;
    unsigned long long* accum = (unsigned long long*)((char*)d_ws + 2048);

    kmeans_init<<<1, 256, 0, stream>>>(x, cent, accum);

    dim3 grid(128, NBATCH), block(256);
    for (int it = 0; it < 10; ++it) {
        kmeans_assign<false><<<grid, block, 0, stream>>>(x, cent, accum, nullptr);
        kmeans_update<<<1, 192, 0, stream>>>(cent, accum);
    }
    kmeans_assign<true><<<grid, block, 0, stream>>>(x, cent, accum, out);
}